// PreCrimeModel_16209206575619
// MI455X (gfx1250) — compile-verified
//
#include <hip/hip_runtime.h>
#include <hip/hip_bf16.h>

// ---------------------------------------------------------------------------
// Hetero-GraphSAGE (2 layers) for MI455X / gfx1250, wave32.
//
// mean_j(x_j) @ Wl == mean_j(x_j @ Wl): project node features through all
// per-layer weights FIRST with V_WMMA_F32_16X16X4_F32 (exact f32), so the
// bandwidth-bound edge scatter moves 16 floats/edge instead of 64.
// Projected features + accumulators are L2-resident (192 MB L2); scatter uses
// GLOBAL_ATOMIC_ADD_F32 which resolves at L2. Degree counts are computed once
// and shared by both layers.
// ---------------------------------------------------------------------------

#define NP 200000
#define NU 50000
#define NE 2000000

typedef __attribute__((ext_vector_type(2))) float v2f;
typedef __attribute__((ext_vector_type(8))) float v8f;

// ---------------------------------------------------------------------------
// C[M x N] = A[M x K] @ B[K x N], f32, row-major. One wave per 16x16 C tile.
// K compile-time (16/32/64) -> fully unrolled straight-line WMMA stream.
// N multiple of 16.
//
// A 16x4 layout (ISA 7.12.2): lanes 0-15 -> M=lane, VGPR0=K+0, VGPR1=K+1;
//                             lanes 16-31 -> M=lane-16, VGPR0=K+2, VGPR1=K+3.
// B 4x16 symmetric:           lanes 0-15 -> N=lane, K+0/K+1;
//                             lanes 16-31 -> N=lane-16, K+2/K+3.
// C/D: VGPR i, lanes 0-15 -> (M=i, N=lane); lanes 16-31 -> (M=i+8, N=lane-16).
//
// Out-of-range rows are handled by clamping the address (stays in-bounds) and
// multiplying the loaded A values by a 0/1 mask: no divergent loads, EXEC
// remains all-ones through the WMMA stream as the ISA requires.
// ---------------------------------------------------------------------------
template <int K>
__global__ void gemm_wmma_f32(const float* __restrict__ A,
                              const float* __restrict__ B,
                              float* __restrict__ C,
                              int M, int N) {
    const int lane = threadIdx.x;        // 0..31 (wave32)
    const int half = lane >> 4;          // 0 or 1
    const int l    = lane & 15;
    const int m0   = blockIdx.x * 16;
    const int n0   = blockIdx.y * 16;

    int rowA = m0 + l;
    const float mask = (rowA < M) ? 1.0f : 0.0f;
    if (rowA >= M) rowA = M - 1;         // clamp: address always valid

    const float* __restrict__ Arow = A + (size_t)rowA * K + half * 2;
    const float* __restrict__ Bcol = B + (size_t)(half * 2) * N + n0 + l;

    v8f acc = {};
#pragma unroll
    for (int k = 0; k < K; k += 4) {
        float2 af = *reinterpret_cast<const float2*>(Arow + k);   // global_load_b64
        v2f a, b;
        a.x = af.x * mask;
        a.y = af.y * mask;
        b.x = Bcol[(size_t)k * N];
        b.y = Bcol[(size_t)(k + 1) * N];
        acc = __builtin_amdgcn_wmma_f32_16x16x4_f32(
            /*neg_a=*/false, a, /*neg_b=*/false, b,
            /*c_mod=*/(short)0, acc, /*reuse_a=*/false, /*reuse_b=*/false);
    }
#pragma unroll
    for (int i = 0; i < 8; ++i) {
        const int row = m0 + i + half * 8;
        if (row < M) C[(size_t)row * N + n0 + l] = acc[i];
    }
}

// Pack a K x 16 weight into column block [colOff, colOff+16) of dst (ld = ldN).
__global__ void pack_weight(float* __restrict__ dst, const float* __restrict__ src,
                            int K, int ldN, int colOff) {
    int idx = blockIdx.x * blockDim.x + threadIdx.x;
    if (idx >= K * 16) return;
    int k = idx >> 4, n = idx & 15;
    dst[(size_t)k * ldN + colOff + n] = src[idx];
}

// Per-destination in-degree (float, reused by both layers).
__global__ void count_edges(const int* __restrict__ dst, float* __restrict__ cnt, int E) {
    int e = blockIdx.x * blockDim.x + threadIdx.x;
    if (e >= E) return;
    atomicAdd(cnt + dst[e], 1.0f);
}

// Scatter 16 projected floats per edge: acc[dst] += feat[src, colOff:colOff+16].
__global__ void scatter_edges(const int* __restrict__ src, const int* __restrict__ dst,
                              const float* __restrict__ feat, int ldF, int colOff,
                              float* __restrict__ acc, int E) {
    int e = blockIdx.x * blockDim.x + threadIdx.x;
    if (e >= E) return;
    const int s = src[e];
    const int d = dst[e];
    const float4* f = reinterpret_cast<const float4*>(feat + (size_t)s * ldF + colOff);
    float* a = acc + (size_t)d * 16;
#pragma unroll
    for (int q = 0; q < 4; ++q) {
        float4 v = f[q];                 // global_load_b128
        atomicAdd(a + 4 * q + 0, v.x);
        atomicAdd(a + 4 * q + 1, v.y);
        atomicAdd(a + 4 * q + 2, v.z);
        atomicAdd(a + 4 * q + 3, v.w);
    }
}

// out = relu(acc/max(cnt,1) + Cdst[:, colOff:+16] + bias)   (single edge type)
__global__ void finalize_one(const float* __restrict__ acc, const float* __restrict__ cnt,
                             const float* __restrict__ Cdst, int ldC, int colOff,
                             const float* __restrict__ bias,
                             float* __restrict__ out, int n) {
    int idx = blockIdx.x * blockDim.x + threadIdx.x;
    if (idx >= n * 16) return;
    int i = idx >> 4, j = idx & 15;
    float c = fmaxf(cnt[i], 1.0f);
    float v = acc[idx] / c + Cdst[(size_t)i * ldC + colOff + j] + bias[j];
    out[idx] = fmaxf(v, 0.0f);
}

// out = relu(sum of two SAGE convs sharing the same destination node set)
__global__ void finalize_two(const float* __restrict__ accA, const float* __restrict__ cntA,
                             const float* __restrict__ accB, const float* __restrict__ cntB,
                             const float* __restrict__ Cdst, int ldC, int colOffA, int colOffB,
                             const float* __restrict__ biasA, const float* __restrict__ biasB,
                             float* __restrict__ out, int n) {
    int idx = blockIdx.x * blockDim.x + threadIdx.x;
    if (idx >= n * 16) return;
    int i = idx >> 4, j = idx & 15;
    float ca = fmaxf(cntA[i], 1.0f);
    float cb = fmaxf(cntB[i], 1.0f);
    float v = accA[idx] / ca + Cdst[(size_t)i * ldC + colOffA + j] + biasA[j]
            + accB[idx] / cb + Cdst[(size_t)i * ldC + colOffB + j] + biasB[j];
    out[idx] = fmaxf(v, 0.0f);
}

// ---------------------------------------------------------------------------

extern "C" void kernel_launch(void* const* d_in, const int* in_sizes, int n_in,
                              void* d_out, int out_size, void* d_ws, size_t ws_size,
                              hipStream_t stream) {
    const float* xp = (const float*)d_in[0];
    const float* xu = (const float*)d_in[1];
    const int* ev_src = (const int*)d_in[2];
    const int* ev_dst = (const int*)d_in[3];
    const int* er_src = (const int*)d_in[4];
    const int* er_dst = (const int*)d_in[5];
    const int* ec_src = (const int*)d_in[6];
    const int* ec_dst = (const int*)d_in[7];
    const float* W1v_l = (const float*)d_in[8];
    const float* W1v_r = (const float*)d_in[9];
    const float* b1v   = (const float*)d_in[10];
    const float* W1r_l = (const float*)d_in[11];
    const float* W1r_r = (const float*)d_in[12];
    const float* b1r   = (const float*)d_in[13];
    const float* W1c_l = (const float*)d_in[14];
    const float* W1c_r = (const float*)d_in[15];
    const float* b1c   = (const float*)d_in[16];
    const float* W2v_l = (const float*)d_in[17];
    const float* W2v_r = (const float*)d_in[18];
    const float* b2v   = (const float*)d_in[19];
    const float* W2r_l = (const float*)d_in[20];
    const float* W2r_r = (const float*)d_in[21];
    const float* b2r   = (const float*)d_in[22];
    const float* W2c_l = (const float*)d_in[23];
    const float* W2c_r = (const float*)d_in[24];
    const float* b2c   = (const float*)d_in[25];

    float* out = (float*)d_out;           // p2 [NP*16] then u2 [NU*16]
    float* ws  = (float*)d_ws;

    // ---- workspace layout (float offsets, 64-float = 256 B aligned) ----
    size_t o = 0;
    auto take = [&](size_t n) { size_t r = o; o += (n + 63) & ~(size_t)63; return r; };
    const size_t oBp1   = take(64 * 64);
    const size_t oBu1   = take(32 * 32);
    const size_t oBp2   = take(16 * 64);
    const size_t oBu2   = take(16 * 32);
    const size_t oCntEv = take(NU);
    const size_t oCntEr = take(NP);
    const size_t oCntEc = take(NP);
    const size_t oAccEv = take((size_t)NU * 16);
    const size_t oAccEr = take((size_t)NP * 16);
    const size_t oAccEc = take((size_t)NP * 16);
    const size_t oCp    = take((size_t)NP * 64);   // reused for layer 1 and 2
    const size_t oCu    = take((size_t)NU * 32);
    const size_t oP1    = take((size_t)NP * 16);
    const size_t oU1    = take((size_t)NU * 16);

    const size_t zero0Bytes   = (oAccEc + (size_t)NP * 16 - oCntEv) * sizeof(float);
    const size_t zeroAccBytes = (oAccEc + (size_t)NP * 16 - oAccEv) * sizeof(float);

    const int TB = 256;
    const int gE   = (NE + TB - 1) / TB;
    const int gP16 = (NP * 16 + TB - 1) / TB;
    const int gU16 = (NU * 16 + TB - 1) / TB;

    // ---- zero counts + accumulators (contiguous region) ----
    hipMemsetAsync(ws + oCntEv, 0, zero0Bytes, stream);

    // ---- degree counts (shared by both layers) ----
    count_edges<<<gE, TB, 0, stream>>>(ev_dst, ws + oCntEv, NE);
    count_edges<<<gE, TB, 0, stream>>>(er_dst, ws + oCntEr, NE);
    count_edges<<<gE, TB, 0, stream>>>(ec_dst, ws + oCntEc, NE);

    // ---- pack concatenated weight matrices ----
    // Bp1 (64x64): [W1v_l | W1c_l | W1r_r | W1c_r]
    pack_weight<<<(64 * 16 + TB - 1) / TB, TB, 0, stream>>>(ws + oBp1, W1v_l, 64, 64, 0);
    pack_weight<<<(64 * 16 + TB - 1) / TB, TB, 0, stream>>>(ws + oBp1, W1c_l, 64, 64, 16);
    pack_weight<<<(64 * 16 + TB - 1) / TB, TB, 0, stream>>>(ws + oBp1, W1r_r, 64, 64, 32);
    pack_weight<<<(64 * 16 + TB - 1) / TB, TB, 0, stream>>>(ws + oBp1, W1c_r, 64, 64, 48);
    // Bu1 (32x32): [W1r_l | W1v_r]
    pack_weight<<<(32 * 16 + TB - 1) / TB, TB, 0, stream>>>(ws + oBu1, W1r_l, 32, 32, 0);
    pack_weight<<<(32 * 16 + TB - 1) / TB, TB, 0, stream>>>(ws + oBu1, W1v_r, 32, 32, 16);
    // Bp2 (16x64): [W2v_l | W2c_l | W2r_r | W2c_r]
    pack_weight<<<(16 * 16 + TB - 1) / TB, TB, 0, stream>>>(ws + oBp2, W2v_l, 16, 64, 0);
    pack_weight<<<(16 * 16 + TB - 1) / TB, TB, 0, stream>>>(ws + oBp2, W2c_l, 16, 64, 16);
    pack_weight<<<(16 * 16 + TB - 1) / TB, TB, 0, stream>>>(ws + oBp2, W2r_r, 16, 64, 32);
    pack_weight<<<(16 * 16 + TB - 1) / TB, TB, 0, stream>>>(ws + oBp2, W2c_r, 16, 64, 48);
    // Bu2 (16x32): [W2r_l | W2v_r]
    pack_weight<<<(16 * 16 + TB - 1) / TB, TB, 0, stream>>>(ws + oBu2, W2r_l, 16, 32, 0);
    pack_weight<<<(16 * 16 + TB - 1) / TB, TB, 0, stream>>>(ws + oBu2, W2v_r, 16, 32, 16);

    // ---- layer 1 dense projections (WMMA) ----
    // Cp = xp @ Bp1 : cols [0:16)=xp@W1v_l, [16:32)=xp@W1c_l, [32:48)=xp@W1r_r, [48:64)=xp@W1c_r
    {
        dim3 grid((NP + 15) / 16, 64 / 16);
        gemm_wmma_f32<64><<<grid, 32, 0, stream>>>(xp, ws + oBp1, ws + oCp, NP, 64);
    }
    // Cu = xu @ Bu1 : cols [0:16)=xu@W1r_l, [16:32)=xu@W1v_r
    {
        dim3 grid((NU + 15) / 16, 32 / 16);
        gemm_wmma_f32<32><<<grid, 32, 0, stream>>>(xu, ws + oBu1, ws + oCu, NU, 32);
    }

    // ---- layer 1 edge scatter (projected 16-float messages) ----
    scatter_edges<<<gE, TB, 0, stream>>>(ev_src, ev_dst, ws + oCp, 64, 0,  ws + oAccEv, NE);
    scatter_edges<<<gE, TB, 0, stream>>>(er_src, er_dst, ws + oCu, 32, 0,  ws + oAccEr, NE);
    scatter_edges<<<gE, TB, 0, stream>>>(ec_src, ec_dst, ws + oCp, 64, 16, ws + oAccEc, NE);

    // ---- layer 1 finalize: mean + dst linear + bias + relu ----
    finalize_one<<<gU16, TB, 0, stream>>>(ws + oAccEv, ws + oCntEv, ws + oCu, 32, 16,
                                          b1v, ws + oU1, NU);
    finalize_two<<<gP16, TB, 0, stream>>>(ws + oAccEr, ws + oCntEr, ws + oAccEc, ws + oCntEc,
                                          ws + oCp, 64, 32, 48, b1r, b1c, ws + oP1, NP);

    // ---- layer 2 ----
    hipMemsetAsync(ws + oAccEv, 0, zeroAccBytes, stream);

    {
        dim3 grid((NP + 15) / 16, 64 / 16);
        gemm_wmma_f32<16><<<grid, 32, 0, stream>>>(ws + oP1, ws + oBp2, ws + oCp, NP, 64);
    }
    {
        dim3 grid((NU + 15) / 16, 32 / 16);
        gemm_wmma_f32<16><<<grid, 32, 0, stream>>>(ws + oU1, ws + oBu2, ws + oCu, NU, 32);
    }

    scatter_edges<<<gE, TB, 0, stream>>>(ev_src, ev_dst, ws + oCp, 64, 0,  ws + oAccEv, NE);
    scatter_edges<<<gE, TB, 0, stream>>>(er_src, er_dst, ws + oCu, 32, 0,  ws + oAccEr, NE);
    scatter_edges<<<gE, TB, 0, stream>>>(ec_src, ec_dst, ws + oCp, 64, 16, ws + oAccEc, NE);

    // ---- layer 2 finalize straight into d_out: p2 first, then u2 ----
    finalize_two<<<gP16, TB, 0, stream>>>(ws + oAccEr, ws + oCntEr, ws + oAccEc, ws + oCntEc,
                                          ws + oCp, 64, 32, 48, b2r, b2c, out, NP);
    finalize_one<<<gU16, TB, 0, stream>>>(ws + oAccEv, ws + oCntEv, ws + oCu, 32, 16,
                                          b2v, out + (size_t)NP * 16, NU);
}